// GCN_Optimizer_86552180949235
// MI455X (gfx1250) — compile-verified
//
#include <hip/hip_runtime.h>
#include <hip/hip_bf16.h>
#include <stdint.h>

typedef __attribute__((ext_vector_type(2))) float v2f;
typedef __attribute__((ext_vector_type(8))) float v8f;

#define HIDDEN      128
#define NUM_CLASSES 5
#define NUM_GRAPHS  128
#define MTILES      4      // 16-row M tiles per block in the WMMA GEMM

// ---------------------------------------------------------------------------
// Degree / normalization: deg[i] = 1 (self loop) + #edges with dst==i;
// dis[i] = rsqrt(deg[i])  (deg >= 1 always, matches reference's where(deg>0)).
// ---------------------------------------------------------------------------
__global__ void k_fill1(float* __restrict__ p, int n) {
    int i = blockIdx.x * blockDim.x + threadIdx.x;
    if (i < n) p[i] = 1.0f;
}

__global__ void k_deg_edges(const long long* __restrict__ dst, float* __restrict__ deg, int E) {
    int e = blockIdx.x * blockDim.x + threadIdx.x;
    if (e < E) atomicAdd(&deg[(int)dst[e]], 1.0f);
}

__global__ void k_rsqrt(float* __restrict__ p, int n) {
    int i = blockIdx.x * blockDim.x + threadIdx.x;
    if (i < n) p[i] = rsqrtf(p[i]);
}

// ---------------------------------------------------------------------------
// Layer 1 GEMM: t[N,128] = x[N,3] @ W1[3,128]   (trivial FLOPs, scalar kernel)
// ---------------------------------------------------------------------------
__global__ __launch_bounds__(256) void k_gemm_k3(const float* __restrict__ x,
                                                 const float* __restrict__ W,
                                                 float* __restrict__ t, int N) {
    int id = blockIdx.x * blockDim.x + threadIdx.x;
    if (id >= N * HIDDEN) return;
    int n = id >> 7, c = id & 127;
    const float* xr = x + (size_t)n * 3;
    t[id] = xr[0] * W[c] + xr[1] * W[HIDDEN + c] + xr[2] * W[2 * HIDDEN + c];
}

// ---------------------------------------------------------------------------
// Layers 2/3 GEMM: C[N,128] = A[N,128] @ B[128,128], fp32 WMMA 16x16x4.
// Block = 256 threads = 8 wave32. Each block computes MTILES (4) 16-row
// M-tiles x 128 cols; wave w owns cols [16w,16w+16). B (64 KB) is staged in
// LDS once per block via CDNA5 async global->LDS DMA (ASYNCcnt-tracked).
// Each B fragment feeds 4 independent accumulator chains (fills XDL pipe).
//
// A-fragment (16x4 f32, ISA 7.12.2): lanes 0-15 -> rows M=0..15 holding
// {K=0,K=1}; lanes 16-31 -> same rows holding {K=2,K=3}.
// B-fragment (4x16 f32): mirror layout (K in VGPR/lane-half, N=lane&15).
// D (16x16 f32): VGPR r: lanes 0-15 -> M=r, lanes 16-31 -> M=r+8.
// ---------------------------------------------------------------------------
__global__ __launch_bounds__(256) void k_gemm_k128(const float* __restrict__ A,
                                                   const float* __restrict__ B,
                                                   float* __restrict__ C, int N) {
    __shared__ float sB[HIDDEN * HIDDEN];  // 64 KB

    // Async DMA stage of B into LDS: 16 B per lane per issue, no VGPR staging.
    // LDS byte address = low 32 bits of the generic pointer (ISA 10.2).
    {
        unsigned           lbase = (unsigned)(size_t)sB;
        unsigned long long gbase = (unsigned long long)(size_t)B;
        for (int i = threadIdx.x; i < HIDDEN * HIDDEN / 4; i += 256) {
            unsigned           laddr = lbase + (unsigned)i * 16u;
            unsigned long long gaddr = gbase + (unsigned long long)i * 16ull;
            asm volatile("global_load_async_to_lds_b128 %0, %1, off"
                         :: "v"(laddr), "v"(gaddr) : "memory");
        }
        asm volatile("s_wait_asynccnt 0x0" ::: "memory");
    }
    __syncthreads();

    const int wave = threadIdx.x >> 5;
    const int lane = threadIdx.x & 31;
    const int col0 = wave << 4;
    const int l15  = lane & 15;
    const int hi   = lane >> 4;        // 0: lanes 0-15, 1: lanes 16-31
    const int koff = hi << 1;          // K sub-offset 0 or 2
    const int col  = col0 + l15;

    const int ntiles = N >> 4;                 // 3125 16-row tiles
    const int mt0    = blockIdx.x * MTILES;

    v8f          acc[MTILES];
    const float* arow[MTILES];
    #pragma unroll
    for (int rt = 0; rt < MTILES; ++rt) {
        acc[rt] = (v8f){0.f, 0.f, 0.f, 0.f, 0.f, 0.f, 0.f, 0.f};
        int mt = mt0 + rt;
        if (mt >= ntiles) mt = ntiles - 1;     // clamp loads; stores guarded below
        arow[rt] = A + ((size_t)(mt << 4) + l15) * HIDDEN;
    }

    for (int kk = 0; kk < HIDDEN; kk += 4) {
        v2f b;
        b.x = sB[(kk + koff)     * HIDDEN + col];
        b.y = sB[(kk + koff + 1) * HIDDEN + col];
        #pragma unroll
        for (int rt = 0; rt < MTILES; ++rt) {
            v2f a;
            a.x = arow[rt][kk + koff];
            a.y = arow[rt][kk + koff + 1];
            acc[rt] = __builtin_amdgcn_wmma_f32_16x16x4_f32(false, a, false, b,
                                                            (short)0, acc[rt],
                                                            false, false);
        }
    }

    #pragma unroll
    for (int rt = 0; rt < MTILES; ++rt) {
        int mt = mt0 + rt;
        if (mt >= ntiles) break;
        int rbase = (mt << 4) + (hi << 3);
        #pragma unroll
        for (int r = 0; r < 8; ++r)
            C[(size_t)(rbase + r) * HIDDEN + col] = acc[rt][r];
    }
}

// ---------------------------------------------------------------------------
// Aggregation: agg = scatter_add(t[src]*dis[src]*dis[dst]) + self-loop term.
// Self-loop pass also serves as the zero-init (writes every element).
// ---------------------------------------------------------------------------
__global__ __launch_bounds__(256) void k_self_init(const float* __restrict__ t,
                                                   const float* __restrict__ dis,
                                                   float* __restrict__ agg, int N) {
    int id = blockIdx.x * blockDim.x + threadIdx.x;   // over N*32 float4 slots
    if (id >= N * (HIDDEN / 4)) return;
    int i = id >> 5;
    float w = dis[i] * dis[i];
    float4 v = ((const float4*)t)[id];
    v.x *= w; v.y *= w; v.z *= w; v.w *= w;
    ((float4*)agg)[id] = v;
}

// One wave32 per edge: 128-bit gather per lane, 4 f32 atomics per lane.
__global__ __launch_bounds__(256) void k_edge_agg(const float* __restrict__ t,
                                                  const long long* __restrict__ src,
                                                  const long long* __restrict__ dst,
                                                  const float* __restrict__ dis,
                                                  float* __restrict__ agg, int E) {
    int e = blockIdx.x * 8 + (threadIdx.x >> 5);
    if (e >= E) return;
    int lane = threadIdx.x & 31;
    int s = (int)src[e], d = (int)dst[e];
    float w = dis[s] * dis[d];
    float4 v = *(const float4*)(t + (size_t)s * HIDDEN + lane * 4);
    float* ag = agg + (size_t)d * HIDDEN + lane * 4;
    atomicAdd(ag + 0, v.x * w);
    atomicAdd(ag + 1, v.y * w);
    atomicAdd(ag + 2, v.z * w);
    atomicAdd(ag + 3, v.w * w);
}

__global__ __launch_bounds__(256) void k_bias_relu(float* __restrict__ h,
                                                   const float* __restrict__ b, int N) {
    int id = blockIdx.x * blockDim.x + threadIdx.x;   // over N*32 float4 slots
    if (id >= N * (HIDDEN / 4)) return;
    float4 v  = ((const float4*)h)[id];
    float4 bb = ((const float4*)b)[id & 31];
    v.x = fmaxf(v.x + bb.x, 0.f);
    v.y = fmaxf(v.y + bb.y, 0.f);
    v.z = fmaxf(v.z + bb.z, 0.f);
    v.w = fmaxf(v.w + bb.w, 0.f);
    ((float4*)h)[id] = v;
}

// ---------------------------------------------------------------------------
// Global mean pool + final linear [G,128] @ [128,5] + bl
// ---------------------------------------------------------------------------
__global__ void k_pool_init(float* __restrict__ pooled, float* __restrict__ cnts) {
    int id = blockIdx.x * blockDim.x + threadIdx.x;
    if (id < NUM_GRAPHS * HIDDEN) pooled[id] = 0.0f;
    if (id < NUM_GRAPHS) cnts[id] = 0.0f;
}

__global__ __launch_bounds__(256) void k_pool(const float* __restrict__ h,
                                              const long long* __restrict__ batch,
                                              float* __restrict__ pooled,
                                              float* __restrict__ cnts, int N) {
    int id = blockIdx.x * blockDim.x + threadIdx.x;   // over N*32 float4 slots
    if (id >= N * (HIDDEN / 4)) return;
    int i = id >> 5, c4 = id & 31;
    int g = (int)batch[i];
    float4 v = ((const float4*)h)[id];
    float* pg = pooled + g * HIDDEN + c4 * 4;
    atomicAdd(pg + 0, v.x);
    atomicAdd(pg + 1, v.y);
    atomicAdd(pg + 2, v.z);
    atomicAdd(pg + 3, v.w);
    if (c4 == 0) atomicAdd(&cnts[g], 1.0f);
}

__global__ void k_final(const float* __restrict__ pooled, const float* __restrict__ cnts,
                        const float* __restrict__ Wl, const float* __restrict__ bl,
                        float* __restrict__ out) {
    int id = blockIdx.x * blockDim.x + threadIdx.x;
    if (id >= NUM_GRAPHS * NUM_CLASSES) return;
    int g = id / NUM_CLASSES, k = id % NUM_CLASSES;
    float inv = 1.0f / fmaxf(cnts[g], 1.0f);
    float s = bl[k];
    const float* pg = pooled + g * HIDDEN;
    #pragma unroll 4
    for (int c = 0; c < HIDDEN; ++c) s += pg[c] * inv * Wl[c * NUM_CLASSES + k];
    out[id] = s;
}

// ---------------------------------------------------------------------------
extern "C" void kernel_launch(void* const* d_in, const int* in_sizes, int n_in,
                              void* d_out, int out_size, void* d_ws, size_t ws_size,
                              hipStream_t stream) {
    const float*     x   = (const float*)d_in[0];
    const long long* ei  = (const long long*)d_in[1];
    const long long* bat = (const long long*)d_in[2];
    const float* W1 = (const float*)d_in[3];
    const float* b1 = (const float*)d_in[4];
    const float* W2 = (const float*)d_in[5];
    const float* b2 = (const float*)d_in[6];
    const float* W3 = (const float*)d_in[7];
    const float* b3 = (const float*)d_in[8];
    const float* Wl = (const float*)d_in[9];
    const float* bl = (const float*)d_in[10];
    float* out = (float*)d_out;

    const int N = in_sizes[0] / 3;      // 50000
    const int E = in_sizes[1] / 2;      // 600000
    const long long* src = ei;
    const long long* dst = ei + E;

    // Workspace carve (256 B aligned)
    size_t off = 0;
    auto carve = [&](size_t bytes) -> float* {
        float* p = (float*)((char*)d_ws + off);
        off = (off + bytes + 255) & ~(size_t)255;
        return p;
    };
    float* dis    = carve((size_t)N * sizeof(float));
    float* pooled = carve((size_t)NUM_GRAPHS * HIDDEN * sizeof(float));
    float* cnts   = carve((size_t)NUM_GRAPHS * sizeof(float));
    float* hA     = carve((size_t)N * HIDDEN * sizeof(float));
    float* hB     = carve((size_t)N * HIDDEN * sizeof(float));

    const int NH4      = N * (HIDDEN / 4);
    const int gNH      = (N * HIDDEN + 255) / 256;
    const int gNH4     = (NH4 + 255) / 256;
    const int gN       = (N + 255) / 256;
    const int gE       = (E + 255) / 256;
    const int gEdge    = (E + 7) / 8;                       // wave per edge
    const int ntiles   = N / 16;                            // 3125
    const int gGemm128 = (ntiles + MTILES - 1) / MTILES;    // 782

    // --- normalization ---
    k_fill1<<<gN, 256, 0, stream>>>(dis, N);
    k_deg_edges<<<gE, 256, 0, stream>>>(dst, dis, E);
    k_rsqrt<<<gN, 256, 0, stream>>>(dis, N);

    // --- layer 1 ---
    k_gemm_k3<<<gNH, 256, 0, stream>>>(x, W1, hA, N);
    k_self_init<<<gNH4, 256, 0, stream>>>(hA, dis, hB, N);
    k_edge_agg<<<gEdge, 256, 0, stream>>>(hA, src, dst, dis, hB, E);
    k_bias_relu<<<gNH4, 256, 0, stream>>>(hB, b1, N);

    // --- layer 2 ---
    k_gemm_k128<<<gGemm128, 256, 0, stream>>>(hB, W2, hA, N);
    k_self_init<<<gNH4, 256, 0, stream>>>(hA, dis, hB, N);
    k_edge_agg<<<gEdge, 256, 0, stream>>>(hA, src, dst, dis, hB, E);
    k_bias_relu<<<gNH4, 256, 0, stream>>>(hB, b2, N);

    // --- layer 3 ---
    k_gemm_k128<<<gGemm128, 256, 0, stream>>>(hB, W3, hA, N);
    k_self_init<<<gNH4, 256, 0, stream>>>(hA, dis, hB, N);
    k_edge_agg<<<gEdge, 256, 0, stream>>>(hA, src, dst, dis, hB, E);
    k_bias_relu<<<gNH4, 256, 0, stream>>>(hB, b3, N);

    // --- pool + classifier ---
    k_pool_init<<<(NUM_GRAPHS * HIDDEN + 255) / 256, 256, 0, stream>>>(pooled, cnts);
    k_pool<<<gNH4, 256, 0, stream>>>(hB, bat, pooled, cnts, N);
    k_final<<<(NUM_GRAPHS * NUM_CLASSES + 255) / 256, 256, 0, stream>>>(pooled, cnts, Wl, bl, out);
    (void)n_in; (void)out_size; (void)ws_size;
}